// SpatialAttention2D_23003844837502
// MI455X (gfx1250) — compile-verified
//
#include <hip/hip_runtime.h>
#include <hip/hip_bf16.h>
#include <math.h>

// Problem constants (B,C,H,W) = (64,1024,32,32)
#define B_DIM   64
#define C_DIM   1024
#define HW      1024
#define SPLITS  32          // hw splits per batch
#define TILE    32          // hw positions per split (SPLITS*TILE == HW)
#define XSTRIDE 34          // padded LDS row stride in floats (even -> 8B aligned, conflict-free)
#define BLOCK   256

typedef float v2f __attribute__((ext_vector_type(2)));
typedef float v8f __attribute__((ext_vector_type(8)));
typedef int   v2i __attribute__((ext_vector_type(2)));

typedef __attribute__((address_space(1))) v2i  g_v2i;
typedef __attribute__((address_space(3))) v2i  l_v2i;
typedef __attribute__((address_space(1))) void g_void;
typedef __attribute__((address_space(3))) void l_void;

#if defined(__has_builtin)
#  if __has_builtin(__builtin_amdgcn_global_load_async_to_lds_b64)
#    define HAVE_ASYNC_LDS 1
#  endif
#  if __has_builtin(__builtin_amdgcn_s_wait_asynccnt)
#    define HAVE_WAIT_ASYNC 1
#  endif
#  if __has_builtin(__builtin_amdgcn_wmma_f32_16x16x4_f32)
#    define HAVE_WMMA_F32X4 1
#  endif
#endif

__device__ __forceinline__ void async_copy_b64(const float* g, float* l) {
#ifdef HAVE_ASYNC_LDS
  g_v2i* gp = (g_v2i*)(g_void*)(void*)const_cast<float*>(g);
  l_v2i* lp = (l_v2i*)(l_void*)(void*)l;
  __builtin_amdgcn_global_load_async_to_lds_b64(gp, lp, 0, 0);
#else
  // Fallback: synchronous copy through VGPRs (both sides 8B aligned).
  float2 v = *(const float2*)g;
  *(float2*)l = v;
#endif
}

__device__ __forceinline__ void wait_async_done() {
#ifdef HAVE_ASYNC_LDS
#  ifdef HAVE_WAIT_ASYNC
  __builtin_amdgcn_s_wait_asynccnt(0);
#  else
  asm volatile("s_wait_asynccnt 0" ::: "memory");
#  endif
#endif
}

// ---------------------------------------------------------------------------
// Kernel 1: one block per (batch, split). Stages x[b, 0:1024, s*32:(s+1)*32]
// into LDS once via async DMA, computes partial softmax stats (m, l) and the
// UNNORMALIZED weighted channel accumulation for this split.
// Single HBM pass over the 256 MB x tensor -> ~11 us roofline at 23.3 TB/s.
// ---------------------------------------------------------------------------
__global__ void __launch_bounds__(BLOCK, 1)
spatial_attn_partial(const float* __restrict__ x,
                     const float* __restrict__ w,
                     const float* __restrict__ bias,
                     float* __restrict__ accBuf,   // [B, SPLITS, C]
                     float* __restrict__ mBuf,     // [B, SPLITS]
                     float* __restrict__ lBuf) {   // [B, SPLITS]
  extern __shared__ float smem[];
  float* xtile  = smem;                       // C_DIM * XSTRIDE
  float* sW     = xtile + C_DIM * XSTRIDE;    // C_DIM
  float* sPart  = sW + C_DIM;                 // 256 partial score sums
  float* sScore = sPart + 256;                // TILE
  float* sP     = sScore + TILE;              // TILE (exp weights)

  const int tid  = threadIdx.x;
  const int bIdx = blockIdx.x >> 5;           // / SPLITS
  const int sIdx = blockIdx.x & (SPLITS - 1);

  const float* xb = x + (size_t)bIdx * C_DIM * HW + (size_t)sIdx * TILE;

  // --- async DMA of the full [1024 x 32] tile into LDS (16384 b64 moves) ---
  // lanes 0..15 cover one row (32 contiguous floats), so each wave issues
  // two contiguous 128B row segments per instruction.
  for (int i = 0; i < 64; ++i) {
    int f  = i * BLOCK + tid;       // pair index, < 16384
    int c  = f >> 4;                // 16 pairs per row
    int t0 = (f & 15) << 1;
    async_copy_b64(xb + (size_t)c * HW + t0, xtile + c * XSTRIDE + t0);
  }
  // stage conv weights while the DMA is in flight
  for (int i = tid; i < C_DIM; i += BLOCK) sW[i] = w[i];
  wait_async_done();
  __syncthreads();

  // --- scores: score[t] = bias + sum_c w[c]*x[c,t] ------------------------
  {
    const int t = tid & 31;         // position within tile (lane id -> no LDS conflicts)
    const int q = tid >> 5;         // channel chunk 0..7 (128 channels each)
    const int cbase = q * 128;
    float acc = 0.f;
#pragma unroll 4
    for (int cc = 0; cc < 128; ++cc) {
      const int c = cbase + cc;
      acc = fmaf(sW[c], xtile[c * XSTRIDE + t], acc);
    }
    sPart[t * 8 + q] = acc;
  }
  __syncthreads();
  if (tid < TILE) {
    float sc = bias[0];
#pragma unroll
    for (int q = 0; q < 8; ++q) sc += sPart[tid * 8 + q];
    sScore[tid] = sc;
  }
  __syncthreads();

  // --- split-local softmax stats ------------------------------------------
  float m = -3.402823466e38f;
#pragma unroll
  for (int t = 0; t < TILE; ++t) m = fmaxf(m, sScore[t]);
  if (tid < TILE) sP[tid] = __expf(sScore[tid] - m);
  __syncthreads();

  if (tid == 0) {
    float l = 0.f;
#pragma unroll
    for (int t = 0; t < TILE; ++t) l += sP[t];
    mBuf[bIdx * SPLITS + sIdx] = m;
    lBuf[bIdx * SPLITS + sIdx] = l;
  }

  // --- unnormalized weighted pooling: acc[c] = sum_t p[t]*x[c,t] ----------
  // thread owns channels {tid, tid+256, tid+512, tid+768}; consecutive lanes
  // read consecutive LDS rows (bank-conflict-free with XSTRIDE=34).
  float* accOut = accBuf + ((size_t)bIdx * SPLITS + sIdx) * C_DIM;
#pragma unroll
  for (int q = 0; q < 4; ++q) {
    const int c = tid + q * BLOCK;
    float acc = 0.f;
#pragma unroll 4
    for (int t = 0; t < TILE; ++t)
      acc = fmaf(sP[t], xtile[c * XSTRIDE + t], acc);
    accOut[c] = acc;    // coalesced store
  }
}

// ---------------------------------------------------------------------------
// Kernel 2: one wave per batch. Merge the 32 splits:
//   M = max_s m_s;  tot = sum_s l_s*e^{m_s-M};  coef_s = e^{m_s-M}/tot
//   out[b,c] = sum_s coef_s * acc_s[c]   -> chained V_WMMA_F32_16X16X4_F32
// (A rows replicate coef over M, B rows are acc[s, c0:c0+16], K = split.)
// ---------------------------------------------------------------------------
__global__ void __launch_bounds__(32)
spatial_attn_combine(const float* __restrict__ accBuf,
                     const float* __restrict__ mBuf,
                     const float* __restrict__ lBuf,
                     float* __restrict__ out) {
  __shared__ float sCoef[SPLITS];
  const int lane = threadIdx.x;
  const int b    = blockIdx.x;
  const float* mb = mBuf + b * SPLITS;
  const float* lb = lBuf + b * SPLITS;

  float M = -3.402823466e38f;
  for (int s = 0; s < SPLITS; ++s) M = fmaxf(M, mb[s]);
  float tot = 0.f;
  for (int s = 0; s < SPLITS; ++s) tot += lb[s] * __expf(mb[s] - M);
  const float inv = 1.0f / tot;
  if (lane < SPLITS) sCoef[lane] = __expf(mb[lane] - M) * inv;
  __syncthreads();

  const float* accB = accBuf + (size_t)b * SPLITS * C_DIM;

#ifdef HAVE_WMMA_F32X4
  // ISA A(16x4 f32) layout: VGPR0 -> K = (lane<16 ? 0 : 2), VGPR1 -> K+1.
  const int half = (lane < 16) ? 0 : 2;
  const int n    = lane & 15;
  for (int j = 0; j < C_DIM / 16; ++j) {
    const int c0 = j * 16;
    v8f d = {};
#pragma unroll
    for (int g = 0; g < SPLITS / 4; ++g) {      // 8 chained K-chunks of 4
      v2f a, bm;
      a.x  = sCoef[g * 4 + half];
      a.y  = sCoef[g * 4 + half + 1];
      bm.x = accB[(size_t)(g * 4 + half)     * C_DIM + c0 + n];
      bm.y = accB[(size_t)(g * 4 + half + 1) * C_DIM + c0 + n];
      d = __builtin_amdgcn_wmma_f32_16x16x4_f32(
              /*neg_a=*/false, a, /*neg_b=*/false, bm,
              /*c_mod=*/(short)0, d, /*reuse_a=*/false, /*reuse_b=*/false);
    }
    // D VGPR0, lanes 0..15 hold row M=0 (all rows identical by construction)
    if (lane < 16) out[b * C_DIM + c0 + lane] = d[0];
  }
#else
  for (int c = lane; c < C_DIM; c += 32) {
    float o = 0.f;
    for (int s = 0; s < SPLITS; ++s) o += sCoef[s] * accB[(size_t)s * C_DIM + c];
    out[b * C_DIM + c] = o;
  }
#endif
}

extern "C" void kernel_launch(void* const* d_in, const int* in_sizes, int n_in,
                              void* d_out, int out_size, void* d_ws, size_t ws_size,
                              hipStream_t stream) {
  const float* x    = (const float*)d_in[0];   // [64,1024,32,32]
  const float* w    = (const float*)d_in[1];   // [1024]
  const float* bias = (const float*)d_in[2];   // scalar
  float* out = (float*)d_out;                  // [64,1024]

  float* accBuf = (float*)d_ws;                              // 64*32*1024 f32 = 8 MB
  float* mBuf   = accBuf + (size_t)B_DIM * SPLITS * C_DIM;   // 2048 f32
  float* lBuf   = mBuf + B_DIM * SPLITS;                     // 2048 f32

  const size_t smemBytes =
      (size_t)(C_DIM * XSTRIDE + C_DIM + 256 + TILE + TILE) * sizeof(float); // ~141 KB

  spatial_attn_partial<<<B_DIM * SPLITS, BLOCK, smemBytes, stream>>>(
      x, w, bias, accBuf, mBuf, lBuf);
  spatial_attn_combine<<<B_DIM, 32, 0, stream>>>(accBuf, mBuf, lBuf, out);
}